// GAT_75204877353212
// MI455X (gfx1250) — compile-verified
//
#include <hip/hip_runtime.h>
#include <math.h>

// ---------------------------------------------------------------------------
// GAT (3-layer, PyG-style) for MI455X / gfx1250, wave32.
// Dense projections use V_WMMA_F32_16X16X4_F32 (true f32 matrix path).
// Edge softmax/scatter phases are atomics-based (memory-bound part).
// ---------------------------------------------------------------------------

typedef float v2f __attribute__((ext_vector_type(2)));
typedef float v8f __attribute__((ext_vector_type(8)));

#define NEG_SLOPE 0.2f
#define BN_EPS 1e-5f

// ---------------- order-preserving float <-> uint encoding (for atomicMax) --
__device__ __forceinline__ unsigned enc_f32(float f) {
    unsigned u = __float_as_uint(f);
    return (u >> 31) ? ~u : (u | 0x80000000u);
}
__device__ __forceinline__ float dec_f32(unsigned e) {
    unsigned u = (e >> 31) ? (e & 0x7fffffffu) : ~e;
    return __uint_as_float(u);
}
__device__ __forceinline__ float lrelu(float x) {
    return x > 0.f ? x : NEG_SLOPE * x;
}
__device__ __forceinline__ void edge_nodes(const int* __restrict__ ei, int E, int e,
                                           int& s, int& d) {
    if (e < E) { s = ei[e]; d = ei[E + e]; }
    else       { s = e - E; d = e - E; }      // appended self-loops
}

// ---------------- zero fill -------------------------------------------------
__global__ void k_fill0(unsigned* __restrict__ p, long long n) {
    long long t = (long long)blockIdx.x * blockDim.x + threadIdx.x;
    if (t < n) p[t] = 0u;
}

// ---------------- WMMA f32 GEMM:  H[N,NCout] = X[N,K] * W[NCout,K]^T --------
// one wave (32 lanes) per 16x16 output tile; K accumulated in steps of 4.
// Branch-free inner loop: out-of-range columns load from a clamped address
// and are zeroed with a select (keeps EXEC all-1s, no saveexec in the loop).
__global__ void k_gemm_wmma(const float* __restrict__ X, const float* __restrict__ W,
                            float* __restrict__ H, int N, int K, int NCout) {
    const int lane = threadIdx.x;          // 0..31
    const int half = lane >> 4;            // ISA 32-bit A/B layout: lane halves hold K pairs
    const int l    = lane & 15;
    const int m0   = blockIdx.x * 16;
    const int n0   = blockIdx.y * 16;

    const int rowL = m0 + l;
    const int rowC = rowL < N ? rowL : (N - 1);     // clamp (EXEC must stay all-1 for WMMA)
    const int col  = n0 + l;
    const bool colOK = col < NCout;
    const int colC = colOK ? col : (NCout - 1);     // clamped load address

    const float* xp = X + (size_t)rowC * K + 2 * half;
    const float* wp = W + (size_t)colC * K + 2 * half;

    v8f c = {};
    for (int k0 = 0; k0 < K; k0 += 4) {
        v2f a, b;
        a.x = xp[k0];
        a.y = xp[k0 + 1];
        float w0 = wp[k0];                           // B[k][n] = W[n][k]
        float w1 = wp[k0 + 1];
        b.x = colOK ? w0 : 0.f;                      // v_cndmask, no EXEC change
        b.y = colOK ? w1 : 0.f;
        c = __builtin_amdgcn_wmma_f32_16x16x4_f32(false, a, false, b,
                                                  (short)0, c, false, false);
    }
    if (colOK) {
        #pragma unroll
        for (int j = 0; j < 8; ++j) {                // D: vgpr j -> M = j + 8*half
            const int row = m0 + j + 8 * half;
            if (row < N) H[(size_t)row * NCout + col] = c[j];
        }
    }
}

// ---------------- per-node attention coefficients ---------------------------
__global__ void k_alphas(const float* __restrict__ H, const float* __restrict__ aS,
                         const float* __restrict__ aD, float* __restrict__ as_,
                         float* __restrict__ ad_, int N, int heads, int C) {
    int t = blockIdx.x * blockDim.x + threadIdx.x;
    if (t >= N * heads) return;
    int n = t / heads, h = t % heads;
    const float* hp = H + (size_t)n * heads * C + (size_t)h * C;
    const float* sv = aS + h * C;
    const float* dv = aD + h * C;
    float s = 0.f, d = 0.f;
    for (int c = 0; c < C; ++c) { float v = hp[c]; s += v * sv[c]; d += v * dv[c]; }
    as_[t] = s; ad_[t] = d;
}

// ---------------- edge pass 1: segment max ----------------------------------
__global__ void k_edge_max(const int* __restrict__ ei, int E, int Etot,
                           const float* __restrict__ as_, const float* __restrict__ ad_,
                           unsigned* __restrict__ segmax, int heads) {
    int t = blockIdx.x * blockDim.x + threadIdx.x;
    if (t >= Etot * heads) return;
    int e = t / heads, h = t % heads;
    int s, d; edge_nodes(ei, E, e, s, d);
    float logit = lrelu(as_[s * heads + h] + ad_[d * heads + h]);
    atomicMax(&segmax[d * heads + h], enc_f32(logit));
}

// ---------------- edge pass 2: segment sum of exp ---------------------------
__global__ void k_edge_sum(const int* __restrict__ ei, int E, int Etot,
                           const float* __restrict__ as_, const float* __restrict__ ad_,
                           const unsigned* __restrict__ segmax, float* __restrict__ segsum,
                           int heads) {
    int t = blockIdx.x * blockDim.x + threadIdx.x;
    if (t >= Etot * heads) return;
    int e = t / heads, h = t % heads;
    int s, d; edge_nodes(ei, E, e, s, d);
    int dh = d * heads + h;
    float logit = lrelu(as_[s * heads + h] + ad_[dh]);
    float ex = expf(logit - dec_f32(segmax[dh]));
    atomicAdd(&segsum[dh], ex);
}

// ---------------- edge pass 3: weighted scatter-add -------------------------
// one thread per (edge, channel); recomputes alpha (cheap vs memory traffic).
__global__ void k_edge_scatter(const int* __restrict__ ei, int E, int Etot,
                               const float* __restrict__ as_, const float* __restrict__ ad_,
                               const unsigned* __restrict__ segmax,
                               const float* __restrict__ segsum,
                               const float* __restrict__ Hlin, float* __restrict__ agg,
                               int heads, int C) {
    const int HC = heads * C;
    long long t = (long long)blockIdx.x * blockDim.x + threadIdx.x;
    if (t >= (long long)Etot * HC) return;
    int e = (int)(t / HC), c = (int)(t % HC);
    int h = c / C;
    int s, d; edge_nodes(ei, E, e, s, d);
    int dh = d * heads + h;
    float logit = lrelu(as_[s * heads + h] + ad_[dh]);
    float ex = expf(logit - dec_f32(segmax[dh]));
    float alpha = ex / (segsum[dh] + 1e-16f);
    atomicAdd(&agg[(size_t)d * HC + c], Hlin[(size_t)s * HC + c] * alpha);
}

// ---------------- bias add + BN statistics ----------------------------------
__global__ void k_bias_stats(float* __restrict__ feat, const float* __restrict__ agg,
                             const float* __restrict__ b, float* __restrict__ sums,
                             int N, int HC) {
    __shared__ float ls[64];
    __shared__ float lq[64];
    int tid = threadIdx.x;
    if (tid < HC) { ls[tid] = 0.f; lq[tid] = 0.f; }
    __syncthreads();
    long long t = (long long)blockIdx.x * blockDim.x + tid;
    if (t < (long long)N * HC) {
        int c = (int)(t % HC);
        float v = agg[t] + b[c];
        feat[t] = v;
        atomicAdd(&ls[c], v);
        atomicAdd(&lq[c], v * v);
    }
    __syncthreads();
    if (tid < HC) {
        atomicAdd(&sums[tid], ls[tid]);
        atomicAdd(&sums[HC + tid], lq[tid]);
    }
}

// ---------------- BN apply (biased variance, matching torch/jnp.var) --------
__global__ void k_bn_apply(float* __restrict__ feat, const float* __restrict__ sums,
                           const float* __restrict__ g, const float* __restrict__ bt,
                           int N, int HC) {
    long long t = (long long)blockIdx.x * blockDim.x + threadIdx.x;
    if (t >= (long long)N * HC) return;
    int c = (int)(t % HC);
    float mean = sums[c] / (float)N;
    float var  = sums[HC + c] / (float)N - mean * mean;
    float sc   = g[c] * rsqrtf(var + BN_EPS);
    feat[t] = (feat[t] - mean) * sc + bt[c];
}

// ---------------- emb = agg + b2 ; log_softmax ; write both outputs ---------
// wave32 per row (8 rows per 256-thread block), shuffle reductions.
__global__ void k_logsoftmax(const float* __restrict__ agg, const float* __restrict__ b,
                             float* __restrict__ logp, float* __restrict__ emb,
                             int N, int C) {
    int wid  = threadIdx.x >> 5;
    int lane = threadIdx.x & 31;
    int row  = blockIdx.x * 8 + wid;
    if (row >= N) return;
    const float* ap = agg + (size_t)row * C;
    float v0 = (lane < C)      ? ap[lane]      + b[lane]      : -INFINITY;
    float v1 = (lane + 32 < C) ? ap[lane + 32] + b[lane + 32] : -INFINITY;
    float mx = fmaxf(v0, v1);
    for (int o = 16; o > 0; o >>= 1) mx = fmaxf(mx, __shfl_xor(mx, o, 32));
    float s = ((lane < C) ? expf(v0 - mx) : 0.f) + ((lane + 32 < C) ? expf(v1 - mx) : 0.f);
    for (int o = 16; o > 0; o >>= 1) s += __shfl_xor(s, o, 32);
    float lse = mx + logf(s);
    if (lane < C) {
        emb [(size_t)row * C + lane] = v0;
        logp[(size_t)row * C + lane] = v0 - lse;
    }
    if (lane + 32 < C) {
        emb [(size_t)row * C + lane + 32] = v1;
        logp[(size_t)row * C + lane + 32] = v1 - lse;
    }
}

// ---------------------------------------------------------------------------
static inline int ceil_div(long long a, long long b) { return (int)((a + b - 1) / b); }

struct LayerBufs {
    float*    hlin;   float* feat;  float* as_;  float* ad_;
    float*    agg;    unsigned* segmax;  float* segsum;  float* sums;
    long long zero_words;          // agg..sums contiguous zero region (in u32 words)
};

static void run_gat_layer(const float* feat_in, int K, const int* ei, int E, int N,
                          const float* W, const float* aS, const float* aD,
                          int heads, int C, const LayerBufs& B, hipStream_t stream) {
    const int HC   = heads * C;
    const int Etot = E + N;

    // zero: agg + segmax + segsum + sums (all-zero bit patterns, incl. encoded -inf)
    k_fill0<<<ceil_div(B.zero_words, 256), 256, 0, stream>>>((unsigned*)B.agg, B.zero_words);

    // projection via WMMA
    dim3 ggrid(ceil_div(N, 16), ceil_div(HC, 16));
    k_gemm_wmma<<<ggrid, 32, 0, stream>>>(feat_in, W, B.hlin, N, K, HC);

    // per-node attention logit halves
    k_alphas<<<ceil_div((long long)N * heads, 256), 256, 0, stream>>>(
        B.hlin, aS, aD, B.as_, B.ad_, N, heads, C);

    // segment softmax + scatter
    k_edge_max<<<ceil_div((long long)Etot * heads, 256), 256, 0, stream>>>(
        ei, E, Etot, B.as_, B.ad_, B.segmax, heads);
    k_edge_sum<<<ceil_div((long long)Etot * heads, 256), 256, 0, stream>>>(
        ei, E, Etot, B.as_, B.ad_, B.segmax, B.segsum, heads);
    k_edge_scatter<<<ceil_div((long long)Etot * HC, 256), 256, 0, stream>>>(
        ei, E, Etot, B.as_, B.ad_, B.segmax, B.segsum, B.hlin, B.agg, heads, C);
}

extern "C" void kernel_launch(void* const* d_in, const int* in_sizes, int n_in,
                              void* d_out, int out_size, void* d_ws, size_t ws_size,
                              hipStream_t stream) {
    const float* x      = (const float*)d_in[0];
    const int*   ei     = (const int*)  d_in[1];   // [2, E] int32
    const float* W0     = (const float*)d_in[2];
    const float* aS0    = (const float*)d_in[3];
    const float* aD0    = (const float*)d_in[4];
    const float* b0     = (const float*)d_in[5];
    const float* W1     = (const float*)d_in[6];
    const float* aS1    = (const float*)d_in[7];
    const float* aD1    = (const float*)d_in[8];
    const float* b1     = (const float*)d_in[9];
    const float* W2     = (const float*)d_in[10];
    const float* aS2    = (const float*)d_in[11];
    const float* aD2    = (const float*)d_in[12];
    const float* b2     = (const float*)d_in[13];
    const float* g0     = (const float*)d_in[14];
    const float* bt0    = (const float*)d_in[15];
    const float* g1     = (const float*)d_in[16];
    const float* bt1    = (const float*)d_in[17];

    const int IN    = 128;
    const int HEADS = 4;
    const int HID   = 16;
    const int OUT   = 40;
    const int H     = HEADS * HID;          // 64
    const int N     = in_sizes[0] / IN;     // 50000
    const int E     = in_sizes[1] / 2;      // 800000

    // ----- workspace layout (floats) -----
    float* p = (float*)d_ws;
    LayerBufs B;
    B.hlin   = p;                 p += (size_t)N * H;      // also holds [N,40] for layer 2
    B.feat   = p;                 p += (size_t)N * H;
    B.as_    = p;                 p += (size_t)N * HEADS;
    B.ad_    = p;                 p += (size_t)N * HEADS;
    B.agg    = p;                 p += (size_t)N * H;      // zero region starts here
    B.segmax = (unsigned*)p;      p += (size_t)N * HEADS;
    B.segsum = p;                 p += (size_t)N * HEADS;
    B.sums   = p;                 p += 2 * H;
    B.zero_words = (long long)N * H + (long long)N * HEADS * 2 + 2 * H;

    const long long nH = (long long)N * H;

    // ----- layer 0: IN -> 4x16, concat, +b0, BN0 -----
    run_gat_layer(x, IN, ei, E, N, W0, aS0, aD0, HEADS, HID, B, stream);
    k_bias_stats<<<ceil_div(nH, 256), 256, 0, stream>>>(B.feat, B.agg, b0, B.sums, N, H);
    k_bn_apply  <<<ceil_div(nH, 256), 256, 0, stream>>>(B.feat, B.sums, g0, bt0, N, H);

    // ----- layer 1: H -> 4x16, concat, +b1, BN1 -----
    run_gat_layer(B.feat, H, ei, E, N, W1, aS1, aD1, HEADS, HID, B, stream);
    k_bias_stats<<<ceil_div(nH, 256), 256, 0, stream>>>(B.feat, B.agg, b1, B.sums, N, H);
    k_bn_apply  <<<ceil_div(nH, 256), 256, 0, stream>>>(B.feat, B.sums, g1, bt1, N, H);

    // ----- layer 2: H -> 40, single head, no concat; then log_softmax -----
    run_gat_layer(B.feat, H, ei, E, N, W2, aS2, aD2, 1, OUT, B, stream);
    float* out_logp = (float*)d_out;
    float* out_emb  = out_logp + (size_t)N * OUT;
    k_logsoftmax<<<ceil_div(N, 8), 256, 0, stream>>>(B.agg, b2, out_logp, out_emb, N, OUT);
}